// TransitionDown_70600672411798
// MI455X (gfx1250) — compile-verified
//
#include <hip/hip_runtime.h>
#include <math.h>

typedef float v2f __attribute__((ext_vector_type(2)));
typedef float v8f __attribute__((ext_vector_type(8)));

#define BB    16
#define NN    4096
#define DD    128
#define SS    1024
#define KK    16
#define CIN   131
#define C1    256
#define C2    256
#define MM    (BB * SS * KK)      // 262144
#define CP1   132                 // padded C_in (132 % 64 == 4 -> conflict-free)
#define CP2   260                 // padded C1    (260 % 64 == 4 -> conflict-free)
#define EPSV  1e-5f

// ---------------------------------------------------------------------------
// Kernel 1: farthest point sampling. One block per batch, 1024 threads,
// running min-distances live in registers (4 points/thread).
// ---------------------------------------------------------------------------
__global__ __launch_bounds__(1024) void fps_kernel(const float* __restrict__ xyz,
                                                   int* __restrict__ fps_idx,
                                                   float* __restrict__ new_xyz) {
    int b = blockIdx.x;
    int t = threadIdx.x;
    const float* x = xyz + (size_t)b * NN * 3;

    float dist[4];
#pragma unroll
    for (int i = 0; i < 4; ++i) dist[i] = 1e10f;

    __shared__ float s_val[32];
    __shared__ int   s_idx[32];
    __shared__ int   s_far;
    __shared__ float s_cx, s_cy, s_cz;

    if (t == 0) s_far = 0;

    for (int j = 0; j < SS; ++j) {
        __syncthreads();
        int far = s_far;
        if (t == 0) {
            fps_idx[b * SS + j] = far;
            float cx = x[far * 3 + 0];
            float cy = x[far * 3 + 1];
            float cz = x[far * 3 + 2];
            s_cx = cx; s_cy = cy; s_cz = cz;
            size_t o = ((size_t)b * SS + j) * 3;
            new_xyz[o + 0] = cx; new_xyz[o + 1] = cy; new_xyz[o + 2] = cz;
        }
        __syncthreads();
        float cx = s_cx, cy = s_cy, cz = s_cz;

        float best = -1.0f;
        int   bi   = 0;
#pragma unroll
        for (int i = 0; i < 4; ++i) {
            int n = i * 1024 + t;
            float dx = x[n * 3 + 0] - cx;
            float dy = x[n * 3 + 1] - cy;
            float dz = x[n * 3 + 2] - cz;
            float d  = dx * dx + dy * dy + dz * dz;
            float dn = fminf(dist[i], d);
            dist[i] = dn;
            if (dn > best) { best = dn; bi = n; }   // ascending n -> first max kept
        }
        // wave32 reduce (prefer larger value; on tie prefer smaller index)
        for (int off = 16; off > 0; off >>= 1) {
            float ov = __shfl_down(best, off, 32);
            int   oi = __shfl_down(bi, off, 32);
            if (ov > best || (ov == best && oi < bi)) { best = ov; bi = oi; }
        }
        int lane = t & 31, wv = t >> 5;
        if (lane == 0) { s_val[wv] = best; s_idx[wv] = bi; }
        __syncthreads();
        if (t < 32) {
            best = s_val[t]; bi = s_idx[t];
            for (int off = 16; off > 0; off >>= 1) {
                float ov = __shfl_down(best, off, 32);
                int   oi = __shfl_down(bi, off, 32);
                if (ov > best || (ov == best && oi < bi)) { best = ov; bi = oi; }
            }
            if (t == 0) s_far = bi;
        }
    }
}

// ---------------------------------------------------------------------------
// Kernel 2: kNN (K=16 smallest squared distances). One thread per query,
// xyz tiled through LDS; register-resident sorted list (static indexing).
// ---------------------------------------------------------------------------
__global__ __launch_bounds__(256) void knn_kernel(const float* __restrict__ xyz,
                                                  const float* __restrict__ new_xyz,
                                                  int* __restrict__ knn_idx) {
    __shared__ float sp[256 * 3];
    int t  = threadIdx.x;
    int b  = blockIdx.x >> 2;              // S/256 = 4 blocks per batch
    int sq = ((blockIdx.x & 3) << 8) + t;

    size_t qo = ((size_t)b * SS + sq) * 3;
    float qx = new_xyz[qo + 0], qy = new_xyz[qo + 1], qz = new_xyz[qo + 2];

    float bd[KK];
    int   bi[KK];
#pragma unroll
    for (int j = 0; j < KK; ++j) { bd[j] = 1e30f; bi[j] = 0; }

    for (int tile = 0; tile < NN / 256; ++tile) {
        __syncthreads();
        int n0 = tile * 256;
        size_t po = ((size_t)b * NN + n0 + t) * 3;
        sp[t * 3 + 0] = xyz[po + 0];
        sp[t * 3 + 1] = xyz[po + 1];
        sp[t * 3 + 2] = xyz[po + 2];
        __syncthreads();
        for (int u = 0; u < 256; ++u) {
            float dx = sp[u * 3 + 0] - qx;
            float dy = sp[u * 3 + 1] - qy;
            float dz = sp[u * 3 + 2] - qz;
            float d  = dx * dx + dy * dy + dz * dz;
            if (d < bd[KK - 1]) {
                float cd = d; int ci = n0 + u;
#pragma unroll
                for (int j = 0; j < KK; ++j) {
                    if (cd < bd[j]) {
                        float td = bd[j]; int ti = bi[j];
                        bd[j] = cd; bi[j] = ci;
                        cd = td; ci = ti;
                    }
                }
            }
        }
    }
    size_t base = ((size_t)b * SS + sq) * KK;
#pragma unroll
    for (int j = 0; j < KK; ++j) knn_idx[base + j] = bi[j];
}

// ---------------------------------------------------------------------------
// Kernel 3: fused gather + GEMM1 (f32 WMMA).  y1[o][m] = sum_c W1[o][c]*feat[m][c] + b1[o]
// Block = 256 thr = 8 waves; covers all 256 output channels x 16 m-columns.
// feat tile (16 x 132) gathered once into LDS and shared by all waves.
// ---------------------------------------------------------------------------
__global__ __launch_bounds__(256) void gemm1_kernel(const float* __restrict__ xyz,
                                                    const float* __restrict__ pts,
                                                    const float* __restrict__ W1,
                                                    const float* __restrict__ b1,
                                                    const int* __restrict__ knn_idx,
                                                    const float* __restrict__ new_xyz,
                                                    float* __restrict__ y1) {
    __shared__ float sB[16 * CP1];
    __shared__ int   s_row[16];
    __shared__ float s_q[16 * 3];

    int t = threadIdx.x;
    int m_base = blockIdx.x * 16;

    if (t < 16) {
        int m  = m_base + t;
        int bb = m >> 14;                  // / (S*K)
        int r  = m & 16383;
        int s  = r >> 4;
        int n  = knn_idx[m];
        s_row[t] = bb * NN + n;
        size_t qo = ((size_t)bb * SS + s) * 3;
        s_q[t * 3 + 0] = new_xyz[qo + 0];
        s_q[t * 3 + 1] = new_xyz[qo + 1];
        s_q[t * 3 + 2] = new_xyz[qo + 2];
    }
    __syncthreads();

    for (int i = t; i < 16 * CP1; i += 256) {
        int mi = i / CP1;
        int c  = i - mi * CP1;
        int row = s_row[mi];
        float v;
        if (c < 3)        v = xyz[(size_t)row * 3 + c] - s_q[mi * 3 + c];
        else if (c < CIN) v = pts[(size_t)row * DD + (c - 3)];
        else              v = 0.0f;
        sB[mi * CP1 + c] = v;
    }
    __syncthreads();

    int lane = t & 31;
    int wv   = t >> 5;
    int col  = lane & 15;
    int klo  = (lane < 16) ? 0 : 2;

    int o0 = (2 * wv) * 16;
    int o1 = (2 * wv + 1) * 16;
    const float* w0 = W1 + (size_t)(o0 + col) * CIN;
    const float* w1 = W1 + (size_t)(o1 + col) * CIN;
    const float* bp = &sB[col * CP1];

    v8f acc0 = {};
    v8f acc1 = {};
    for (int cb = 0; cb < CP1; cb += 4) {
        int c0 = cb + klo, c1 = cb + klo + 1;
        float m0 = (c0 < CIN) ? 1.0f : 0.0f;
        float m1 = (c1 < CIN) ? 1.0f : 0.0f;
        int cc0 = c0 < CIN ? c0 : CIN - 1;
        int cc1 = c1 < CIN ? c1 : CIN - 1;
        v2f bv; bv.x = bp[c0]; bv.y = bp[c1];
        v2f a0; a0.x = w0[cc0] * m0; a0.y = w0[cc1] * m1;
        v2f a1; a1.x = w1[cc0] * m0; a1.y = w1[cc1] * m1;
        acc0 = __builtin_amdgcn_wmma_f32_16x16x4_f32(false, a0, false, bv, (short)0, acc0, false, false);
        acc1 = __builtin_amdgcn_wmma_f32_16x16x4_f32(false, a1, false, bv, (short)0, acc1, false, false);
    }

    int mo = m_base + col;
    int hi = (lane < 16) ? 0 : 8;
#pragma unroll
    for (int v = 0; v < 8; ++v) {
        int r0 = o0 + v + hi;
        int r1 = o1 + v + hi;
        y1[(size_t)r0 * MM + mo] = acc0[v] + b1[r0];
        y1[(size_t)r1 * MM + mo] = acc1[v] + b1[r1];
    }
}

// ---------------------------------------------------------------------------
// Kernel 4/6: per-channel BN statistics over a [C][M] activation buffer.
// One block per channel -> mean + rsqrt(var+eps).
// ---------------------------------------------------------------------------
__global__ __launch_bounds__(256) void stats_kernel(const float* __restrict__ y,
                                                    float* __restrict__ mean,
                                                    float* __restrict__ istd) {
    int c = blockIdx.x;
    int t = threadIdx.x;
    const float* row = y + (size_t)c * MM;
    float s = 0.0f, q = 0.0f;
    for (int i = t; i < MM; i += 256) {
        float v = row[i];
        s += v;
        q += v * v;
    }
    for (int off = 16; off > 0; off >>= 1) {
        s += __shfl_down(s, off, 32);
        q += __shfl_down(q, off, 32);
    }
    __shared__ float rs[8], rq[8];
    int lane = t & 31, wv = t >> 5;
    if (lane == 0) { rs[wv] = s; rq[wv] = q; }
    __syncthreads();
    if (t == 0) {
        float S = 0.0f, Q = 0.0f;
#pragma unroll
        for (int i = 0; i < 8; ++i) { S += rs[i]; Q += rq[i]; }
        float mu  = S / (float)MM;
        float var = Q / (float)MM - mu * mu;
        mean[c] = mu;
        istd[c] = rsqrtf(var + EPSV);
    }
}

// ---------------------------------------------------------------------------
// Kernel 5: GEMM2 with BN+ReLU applied to the input while staging into LDS.
// y2[o][m] = sum_c W2[o][c] * relu(bn(y1[c][m])) + b2[o]
// ---------------------------------------------------------------------------
__global__ __launch_bounds__(256) void gemm2_kernel(const float* __restrict__ y1,
                                                    const float* __restrict__ W2,
                                                    const float* __restrict__ b2,
                                                    const float* __restrict__ mean1,
                                                    const float* __restrict__ istd1,
                                                    const float* __restrict__ g1,
                                                    const float* __restrict__ bt1,
                                                    float* __restrict__ y2) {
    __shared__ float sB[16 * CP2];
    int t = threadIdx.x;
    int m_base = blockIdx.x * 16;

    for (int i = t; i < 16 * C1; i += 256) {
        int c  = i >> 4;
        int mi = i & 15;
        float v = y1[(size_t)c * MM + m_base + mi];
        v = (v - mean1[c]) * istd1[c] * g1[c] + bt1[c];
        v = fmaxf(v, 0.0f);
        sB[mi * CP2 + c] = v;
    }
    __syncthreads();

    int lane = t & 31;
    int wv   = t >> 5;
    int col  = lane & 15;
    int klo  = (lane < 16) ? 0 : 2;

    int o0 = (2 * wv) * 16;
    int o1 = (2 * wv + 1) * 16;
    const float* w0 = W2 + (size_t)(o0 + col) * C1;
    const float* w1 = W2 + (size_t)(o1 + col) * C1;
    const float* bp = &sB[col * CP2];

    v8f acc0 = {};
    v8f acc1 = {};
    for (int cb = 0; cb < C1; cb += 4) {
        int c0 = cb + klo, c1 = cb + klo + 1;
        v2f bv; bv.x = bp[c0]; bv.y = bp[c1];
        v2f a0; a0.x = w0[c0]; a0.y = w0[c1];
        v2f a1; a1.x = w1[c0]; a1.y = w1[c1];
        acc0 = __builtin_amdgcn_wmma_f32_16x16x4_f32(false, a0, false, bv, (short)0, acc0, false, false);
        acc1 = __builtin_amdgcn_wmma_f32_16x16x4_f32(false, a1, false, bv, (short)0, acc1, false, false);
    }

    int mo = m_base + col;
    int hi = (lane < 16) ? 0 : 8;
#pragma unroll
    for (int v = 0; v < 8; ++v) {
        int r0 = o0 + v + hi;
        int r1 = o1 + v + hi;
        y2[(size_t)r0 * MM + mo] = acc0[v] + b2[r0];
        y2[(size_t)r1 * MM + mo] = acc1[v] + b2[r1];
    }
}

// ---------------------------------------------------------------------------
// Kernel 7: BN+ReLU on y2 + max over K, write new_points.
// Thread mapping chosen so the dominant reads (y2) are fully coalesced.
// ---------------------------------------------------------------------------
__global__ __launch_bounds__(256) void finalize_kernel(const float* __restrict__ y2,
                                                       const float* __restrict__ mean2,
                                                       const float* __restrict__ istd2,
                                                       const float* __restrict__ g2,
                                                       const float* __restrict__ bt2,
                                                       float* __restrict__ out_points) {
    int t  = blockIdx.x * blockDim.x + threadIdx.x;   // exactly B*S*C2 threads
    int bs = t & (BB * SS - 1);
    int o  = t >> 14;                                  // / (B*S)
    float mu = mean2[o];
    float sc = istd2[o] * g2[o];
    float bt = bt2[o];
    const float* r = y2 + (size_t)o * MM + (size_t)bs * KK;
    float mx = 0.0f;                                   // relu outputs are >= 0
#pragma unroll
    for (int k = 0; k < KK; ++k) {
        float v = fmaxf((r[k] - mu) * sc + bt, 0.0f);
        mx = fmaxf(mx, v);
    }
    out_points[(size_t)bs * C2 + o] = mx;
}

// ---------------------------------------------------------------------------
extern "C" void kernel_launch(void* const* d_in, const int* in_sizes, int n_in,
                              void* d_out, int out_size, void* d_ws, size_t ws_size,
                              hipStream_t stream) {
    const float* xyz = (const float*)d_in[0];
    const float* pts = (const float*)d_in[1];
    const float* W1  = (const float*)d_in[2];
    const float* b1  = (const float*)d_in[3];
    const float* g1  = (const float*)d_in[4];
    const float* bt1 = (const float*)d_in[5];
    const float* W2  = (const float*)d_in[6];
    const float* b2  = (const float*)d_in[7];
    const float* g2  = (const float*)d_in[8];
    const float* bt2 = (const float*)d_in[9];

    float* out_xyz    = (float*)d_out;                       // [B,S,3]
    float* out_points = (float*)d_out + (size_t)BB * SS * 3; // [B,S,C2]

    // workspace layout
    char* ws = (char*)d_ws;
    int*   fps_idx = (int*)ws;                                  // B*S ints
    int*   knn_idx = (int*)(ws + 65536);                        // M ints
    float* y1      = (float*)(ws + 1114112);                    // C1*M f32 (256 MiB)
    float* y2      = (float*)(ws + 1114112 + (size_t)C1 * MM * 4);
    float* statp   = (float*)(ws + 1114112 + 2 * (size_t)C1 * MM * 4);
    float* mean1 = statp, *istd1 = statp + C1;
    float* mean2 = statp + 2 * C1, *istd2 = statp + 3 * C1;

    fps_kernel<<<BB, 1024, 0, stream>>>(xyz, fps_idx, out_xyz);
    knn_kernel<<<BB * (SS / 256), 256, 0, stream>>>(xyz, out_xyz, knn_idx);
    gemm1_kernel<<<MM / 16, 256, 0, stream>>>(xyz, pts, W1, b1, knn_idx, out_xyz, y1);
    stats_kernel<<<C1, 256, 0, stream>>>(y1, mean1, istd1);
    gemm2_kernel<<<MM / 16, 256, 0, stream>>>(y1, W2, b2, mean1, istd1, g1, bt1, y2);
    stats_kernel<<<C2, 256, 0, stream>>>(y2, mean2, istd2);
    finalize_kernel<<<(BB * SS * C2) / 256, 256, 0, stream>>>(y2, mean2, istd2, g2, bt2, out_points);
}